// MultiHeadAttentionPooling_8710193676614
// MI455X (gfx1250) — compile-verified
//
#include <hip/hip_runtime.h>

#define BN 4
#define NN 2048
#define DD 512
#define HH 8
#define SS 2049   // 1 + NN

typedef __attribute__((ext_vector_type(2))) float v2f;
typedef __attribute__((ext_vector_type(8))) float v8f;

// workspace layout (float offsets)
#define OFF_QK   0                    // [16][512]  (rows 8..15 zero), pre-scaled by 1/8
#define OFF_QD   8192                 // [16]       per-head q0.bk constant (scaled)
#define OFF_Z    8208                 // [B][8][512] attn-weighted (unnormalized) feature sums
#define OFF_W    24592                // [B][S][8]   unnormalized exp(score)

__device__ __forceinline__ int imin(int a, int b) { return a < b ? a : b; }

__device__ __forceinline__ float loadXelem(const float* __restrict__ nf,
                                           const float* __restrict__ ct,
                                           int b, int s, int col) {
  int sc = imin(s, SS - 1);
  const float* p = (sc == 0) ? (ct + col)
                             : (nf + ((size_t)b * NN + (sc - 1)) * DD + col);
  return *p;
}

// ---------------------------------------------------------------------------
// K0: q0 slice per head-block, qk[h][e] = (1/8) * sum_{j<64} q0[h*64+j]*Wk[h*64+j][e]
// grid: 16 blocks (one per padded head row), 256 threads
// ---------------------------------------------------------------------------
__global__ __launch_bounds__(256)
void gap_k0_prep(const float* __restrict__ ct, const float* __restrict__ Wq,
                 const float* __restrict__ bq, const float* __restrict__ Wk,
                 const float* __restrict__ bk, float* __restrict__ ws) {
  __shared__ float s_q0[64];
  const int h = blockIdx.x;            // 0..15 (8..15 are zero pad rows)
  const int tid = threadIdx.x;

  if (h < HH && tid < 64) {
    const int d = h * 64 + tid;
    float acc = bq[d];
    const float* wr = Wq + (size_t)d * DD;
    for (int e = 0; e < DD; ++e) acc += ct[e] * wr[e];
    s_q0[tid] = acc;
  }
  __syncthreads();

  for (int e = tid; e < DD; e += 256) {
    float acc = 0.f;
    if (h < HH) {
      const float* wc = Wk + (size_t)(h * 64) * DD + e;
      for (int j = 0; j < 64; ++j) acc += s_q0[j] * wc[(size_t)j * DD];
      acc *= 0.125f;                    // 1/sqrt(HD) folded in
    }
    ws[OFF_QK + (size_t)h * DD + e] = acc;
  }
  if (tid == 0) {
    float acc = 0.f;
    if (h < HH)
      for (int j = 0; j < 64; ++j) acc += s_q0[j] * bk[h * 64 + j];
    ws[OFF_QD + h] = 0.125f * acc;
  }
}

// ---------------------------------------------------------------------------
// K1: scores + exp.  w[b,s,h] = exp( X[b,s]·qk_h + qd_h ) (masked)
// grid: (ceil(S/64), B), 128 threads (4 waves, 16 rows/wave)
// WMMA f32 16x16x4: A = X rows (M=16,K=4), B = qk^T (K=4,N=16 heads)
// ---------------------------------------------------------------------------
__global__ __launch_bounds__(128)
void gap_k1_scores(const float* __restrict__ nf, const float* __restrict__ ct,
                   const float* __restrict__ masks, float* __restrict__ ws) {
  const int b = blockIdx.y;
  const int lane = threadIdx.x & 31;
  const int wave = threadIdx.x >> 5;
  const int r0 = blockIdx.x * 64 + wave * 16;   // first output row of this wave
  const int g = lane >> 4;                      // lane group (0/1)
  const int lm = lane & 15;

  // A-operand row for this lane (M = lane%16), clamped into valid memory
  const int arow = imin(r0 + lm, SS - 1);
  const float* aptr = ((arow == 0) ? ct
                                   : (nf + ((size_t)b * NN + (arow - 1)) * DD)) + 2 * g;
  // B-operand: qk row (N = lane%16)
  const float* qk = ws + OFF_QK + (size_t)lm * DD + 2 * g;

  v8f c0 = {}; v8f c1 = {};
  #pragma unroll 4
  for (int k0 = 0; k0 < DD; k0 += 8) {
    v2f a0 = *(const v2f*)(aptr + k0);
    v2f b0 = *(const v2f*)(qk + k0);
    c0 = __builtin_amdgcn_wmma_f32_16x16x4_f32(false, a0, false, b0,
                                               (short)0, c0, false, false);
    v2f a1 = *(const v2f*)(aptr + k0 + 4);
    v2f b1 = *(const v2f*)(qk + k0 + 4);
    c1 = __builtin_amdgcn_wmma_f32_16x16x4_f32(false, a1, false, b1,
                                               (short)0, c1, false, false);
  }

  // C layout: element r of lane L -> row M = r + 8*(L/16), head N = L%16
  const int head = lm;
  const float qd = ws[OFF_QD + head];
  #pragma unroll
  for (int r = 0; r < 8; ++r) {
    const int si = r0 + r + 8 * g;
    const float sc = c0[r] + c1[r] + qd;
    float mk = 1.f;
    if (si > 0 && si < SS) mk = masks[(size_t)b * NN + (si - 1)];
    if (head < HH && si < SS) {
      const float w = (mk != 0.f) ? __expf(sc) : 0.f;
      ws[OFF_W + ((size_t)b * SS + si) * 8 + head] = w;
    }
  }
}

// ---------------------------------------------------------------------------
// K2: z[b,h,:] = sum_s w[b,s,h] * X[b,s,:]   (unnormalized)
// grid: (D/16, B), 256 threads (8 waves split the S dimension), LDS reduction
// WMMA f32 16x16x4: A = w^T (M=16 heads, K=4 s), B = X (K=4 s, N=16 cols)
// ---------------------------------------------------------------------------
__global__ __launch_bounds__(256)
void gap_k2_accum(const float* __restrict__ nf, const float* __restrict__ ct,
                  float* __restrict__ ws) {
  __shared__ float zred[8][8][16];
  const int b = blockIdx.y;
  const int n0 = blockIdx.x * 16;
  const int lane = threadIdx.x & 31;
  const int wave = threadIdx.x >> 5;
  const int g = lane >> 4;
  const int lm = lane & 15;
  const int head = lm;                 // A operand: M = lane%16 = head
  const int col = n0 + lm;             // B operand: N = lane%16
  const int sBase = wave * 264;        // 8*264 = 2112 >= 2049

  const float* wb = ws + OFF_W + (size_t)b * SS * 8;
  v8f c0 = {}; v8f c1 = {};
  for (int kk = 0; kk < 264; kk += 8) {
    {
      const int s0 = sBase + kk + 2 * g;
      v2f a, x;
      const float w0 = wb[(size_t)imin(s0, SS - 1) * 8 + (head & 7)];
      const float w1 = wb[(size_t)imin(s0 + 1, SS - 1) * 8 + (head & 7)];
      a.x = (head < HH && s0     < SS) ? w0 : 0.f;
      a.y = (head < HH && s0 + 1 < SS) ? w1 : 0.f;
      x.x = loadXelem(nf, ct, b, s0,     col);
      x.y = loadXelem(nf, ct, b, s0 + 1, col);
      c0 = __builtin_amdgcn_wmma_f32_16x16x4_f32(false, a, false, x,
                                                 (short)0, c0, false, false);
    }
    {
      const int s0 = sBase + kk + 4 + 2 * g;
      v2f a, x;
      const float w0 = wb[(size_t)imin(s0, SS - 1) * 8 + (head & 7)];
      const float w1 = wb[(size_t)imin(s0 + 1, SS - 1) * 8 + (head & 7)];
      a.x = (head < HH && s0     < SS) ? w0 : 0.f;
      a.y = (head < HH && s0 + 1 < SS) ? w1 : 0.f;
      x.x = loadXelem(nf, ct, b, s0,     col);
      x.y = loadXelem(nf, ct, b, s0 + 1, col);
      c1 = __builtin_amdgcn_wmma_f32_16x16x4_f32(false, a, false, x,
                                                 (short)0, c1, false, false);
    }
  }

  const v8f c = c0 + c1;
  // C: element r of lane L -> head M = r + 8*(L/16), col N = L%16.
  // Heads 8..15 are exactly zero (their A rows were zero); only g==0 matters.
  if (g == 0) {
    #pragma unroll
    for (int r = 0; r < 8; ++r) zred[wave][r][lm] = c[r];
  }
  __syncthreads();
  if (threadIdx.x < 128) {
    const int h = threadIdx.x >> 4;
    const int cl = threadIdx.x & 15;
    float s = 0.f;
    #pragma unroll
    for (int wv = 0; wv < 8; ++wv) s += zred[wv][h][cl];
    ws[OFF_Z + ((size_t)b * HH + h) * DD + n0 + cl] = s;
  }
}

// ---------------------------------------------------------------------------
// K3: denominators, Wv-GEMV, Wo-GEMV, residual, LayerNorm. grid: B, 512 thr
// ---------------------------------------------------------------------------
__global__ __launch_bounds__(512)
void gap_k3_final(const float* __restrict__ ct, const float* __restrict__ Wv,
                  const float* __restrict__ bv, const float* __restrict__ Wo,
                  const float* __restrict__ bo, const float* __restrict__ gam,
                  const float* __restrict__ bet, const float* __restrict__ ws,
                  float* __restrict__ out) {
  __shared__ float s_red[512];
  __shared__ float s_att[DD];
  __shared__ float s_y[DD];
  __shared__ float s_inv[HH];
  __shared__ float s_mu, s_var;
  const int b = blockIdx.x;
  const int t = threadIdx.x;

  // 1) per-head softmax denominator: sum_s w[b,s,h]
  {
    const int h = t >> 6, j = t & 63;          // 8 heads x 64 threads
    float p = 0.f;
    const float* wb = ws + OFF_W + (size_t)b * SS * 8;
    for (int s = j; s < SS; s += 64) p += wb[(size_t)s * 8 + h];
    s_red[t] = p;
    __syncthreads();
    for (int off = 32; off > 0; off >>= 1) {
      if (j < off) s_red[t] += s_red[t + off];
      __syncthreads();
    }
    if (j == 0) s_inv[h] = 1.f / s_red[t];
    __syncthreads();
  }
  // 2) attention output: att[d] = inv[h] * sum_e z[h][e]*Wv[d][e] + bv[d]
  {
    const int d = t, h = d >> 6;
    const float* zr = ws + OFF_Z + ((size_t)b * HH + h) * DD;
    const float* wr = Wv + (size_t)d * DD;
    float acc = 0.f;
    for (int e = 0; e < DD; ++e) acc += zr[e] * wr[e];
    s_att[d] = acc * s_inv[h] + bv[d];
  }
  __syncthreads();
  // 3) output projection + residual (x row 0 == class_token)
  {
    const int d = t;
    const float* wr = Wo + (size_t)d * DD;
    float acc = bo[d];
    for (int e = 0; e < DD; ++e) acc += s_att[e] * wr[e];
    s_y[d] = acc + ct[d];
  }
  __syncthreads();
  // 4) LayerNorm over D
  const float y = s_y[t];
  s_red[t] = y; __syncthreads();
  for (int off = 256; off > 0; off >>= 1) {
    if (t < off) s_red[t] += s_red[t + off];
    __syncthreads();
  }
  if (t == 0) s_mu = s_red[0] / (float)DD;
  __syncthreads();
  const float dc = y - s_mu;
  s_red[t] = dc * dc; __syncthreads();
  for (int off = 256; off > 0; off >>= 1) {
    if (t < off) s_red[t] += s_red[t + off];
    __syncthreads();
  }
  if (t == 0) s_var = s_red[0] / (float)DD;
  __syncthreads();
  out[(size_t)b * DD + t] = dc * rsqrtf(s_var + 1e-5f) * gam[t] + bet[t];
}

// ---------------------------------------------------------------------------
extern "C" void kernel_launch(void* const* d_in, const int* in_sizes, int n_in,
                              void* d_out, int out_size, void* d_ws, size_t ws_size,
                              hipStream_t stream) {
  (void)in_sizes; (void)n_in; (void)out_size; (void)ws_size;
  const float* nf    = (const float*)d_in[0];   // node_feat   [B,N,D]
  // d_in[1] edge_weights, d_in[2] adj_matrix: provably unused for row-0 output
  const float* masks = (const float*)d_in[3];   // [B,N]
  const float* ct    = (const float*)d_in[4];   // class_token [D]
  const float* Wq    = (const float*)d_in[5];
  const float* bq    = (const float*)d_in[6];
  const float* Wk    = (const float*)d_in[7];
  const float* bk    = (const float*)d_in[8];
  const float* Wv    = (const float*)d_in[9];
  const float* bv    = (const float*)d_in[10];
  const float* Wo    = (const float*)d_in[11];
  const float* bo    = (const float*)d_in[12];
  const float* gam   = (const float*)d_in[13];
  const float* bet   = (const float*)d_in[14];
  float* ws  = (float*)d_ws;
  float* out = (float*)d_out;

  gap_k0_prep<<<dim3(16), dim3(256), 0, stream>>>(ct, Wq, bq, Wk, bk, ws);
  gap_k1_scores<<<dim3((SS + 63) / 64, BN), dim3(128), 0, stream>>>(nf, ct, masks, ws);
  gap_k2_accum<<<dim3(DD / 16, BN), dim3(256), 0, stream>>>(nf, ct, ws);
  gap_k3_final<<<dim3(BN), dim3(512), 0, stream>>>(ct, Wv, bv, Wo, bo, gam, bet, ws, out);
}